// LocalLoss2_73796128080064
// MI455X (gfx1250) — compile-verified
//
#include <hip/hip_runtime.h>
#include <math.h>

// ---------------- problem constants ----------------
#define NIMG 32
#define NCH  3
#define HH   224
#define WW   224
#define NPLANES (NIMG*NCH)       // 96 single-channel planes
#define PLANE   (HH*WW)          // 50176
#define NPIX    ((size_t)NPLANES*PLANE)
#define PI_F 3.14159265358979323846f

// Sobel/magbin tiling
#define TW 32
#define TH 8
#define LW (TW+2)                // 34
#define LH (TH+2)                // 10

#define PARTS 1024               // fixed partial count per reduction (deterministic)

typedef __attribute__((ext_vector_type(2))) float v2f;
typedef __attribute__((ext_vector_type(8))) float v8f;

// ---------------- CDNA5 feature detection ----------------
#if defined(__has_builtin)
#  if __has_builtin(__builtin_amdgcn_global_load_async_to_lds_b32)
#    define ASYNC_LDS_OK 1
#  endif
#  if __has_builtin(__builtin_amdgcn_s_wait_asynccnt)
#    define WAIT_ASYNC_BUILTIN 1
#  endif
#  if __has_builtin(__builtin_amdgcn_wmma_f32_16x16x4_f32)
#    define WMMA_F32_OK 1
#  endif
#endif

__device__ __forceinline__ void ld_async_f32(const float* g, float* l) {
#if defined(ASYNC_LDS_OK)
  __builtin_amdgcn_global_load_async_to_lds_b32(
      (__attribute__((address_space(1))) int*)(g),
      (__attribute__((address_space(3))) int*)(l),
      0, 0);
#else
  *l = *g;
#endif
}

__device__ __forceinline__ void wait_async_lds() {
#if defined(ASYNC_LDS_OK)
#  if defined(WAIT_ASYNC_BUILTIN)
  __builtin_amdgcn_s_wait_asynccnt(0);
#  else
  asm volatile("s_wait_asynccnt 0" ::: "memory");
#  endif
#endif
}

// ---------------- kernel 1: per-pixel (mag, bin) via async-LDS tiled Sobel ----------------
__global__ void magbin_kernel(const float* __restrict__ imgs,
                              float* __restrict__ magp,
                              unsigned char* __restrict__ binp)
{
  __shared__ float tile[LH][LW];

  int b  = blockIdx.x;
  int tX = b % (WW/TW);  int r = b / (WW/TW);
  int tY = r % (HH/TH);  int plane = r / (HH/TH);
  int x0 = tX*TW, y0 = tY*TH;
  const float* img = imgs + (size_t)plane * PLANE;

  for (int i = threadIdx.x; i < LH*LW; i += blockDim.x) {
    int ly = i / LW, lx = i % LW;
    int gy = y0 + ly - 1;
    int gx = x0 + lx - 1;
    gy = gy < 0 ? 1 : (gy >= HH ? HH-2 : gy);   // reflect pad = 1
    gx = gx < 0 ? 1 : (gx >= WW ? WW-2 : gx);
    ld_async_f32(img + (size_t)gy*WW + gx, &tile[ly][lx]);
  }
  wait_async_lds();
  __syncthreads();

  int lx = threadIdx.x % TW;
  int ly = threadIdx.x / TW;

  float p00 = tile[ly  ][lx], p01 = tile[ly  ][lx+1], p02 = tile[ly  ][lx+2];
  float p10 = tile[ly+1][lx],                          p12 = tile[ly+1][lx+2];
  float p20 = tile[ly+2][lx], p21 = tile[ly+2][lx+1], p22 = tile[ly+2][lx+2];

  float gxv = (p00 + 2.f*p10 + p20) - (p02 + 2.f*p12 + p22);
  float gyv = (p00 + 2.f*p01 + p02) - (p20 + 2.f*p21 + p22);

  float mag = sqrtf(gxv*gxv + gyv*gyv);
  float ph  = atan2f(gxv, gyv);
  int   bin = (int)floorf(ph * (9.0f/PI_F));
  bin %= 9; if (bin < 0) bin += 9;

  size_t pidx = (size_t)plane*PLANE + (size_t)(y0+ly)*WW + (x0+lx);
  magp[pidx] = mag;
  binp[pidx] = (unsigned char)bin;
}

// ---------------- kernel 2: mask total sum (stage 1) ----------------
__global__ void mask_reduce_kernel(const float* __restrict__ mask,
                                   float* __restrict__ partial)
{
  __shared__ float red[256];
  const size_t tot = (size_t)NIMG * PLANE;
  float v = 0.f;
  for (size_t i = (size_t)blockIdx.x*blockDim.x + threadIdx.x; i < tot;
       i += (size_t)gridDim.x*blockDim.x)
    v += mask[i];
  red[threadIdx.x] = v; __syncthreads();
  for (int off = 128; off > 0; off >>= 1) {
    if ((int)threadIdx.x < off) red[threadIdx.x] += red[threadIdx.x + off];
    __syncthreads();
  }
  if (threadIdx.x == 0) partial[blockIdx.x] = red[0];
}

// ---------------- kernel 3a: per-scale fused loss, VALU gather path ----------------
template<int S>
__global__ void scale_loss_kernel(const float* __restrict__ magp,
                                  const unsigned char* __restrict__ binp,
                                  const float* __restrict__ mask,
                                  const float* __restrict__ pred,
                                  const float* __restrict__ pw,
                                  const float* __restrict__ pb,
                                  float* __restrict__ partial)
{
  constexpr int Ho = HH / S, Wo = WW / S;
  constexpr int TOT = NIMG * Ho * Wo;

  __shared__ float spw[81];
  __shared__ float spb[9];
  __shared__ float red[256];
  for (int i = threadIdx.x; i < 81; i += blockDim.x) spw[i] = pw[i];
  if (threadIdx.x < 9) spb[threadIdx.x] = pb[threadIdx.x];
  __syncthreads();

  float tsum = 0.f;
  for (int idx = blockIdx.x*blockDim.x + threadIdx.x; idx < TOT;
       idx += gridDim.x*blockDim.x) {
    int ox = idx % Wo; int r = idx / Wo;
    int oy = r % Ho;   int n = r / Ho;

    float acc = 0.f;
    #pragma unroll
    for (int c = 0; c < 3; ++c) {
      size_t pix = (size_t)(n*3 + c)*PLANE + (size_t)(oy*S)*WW + (size_t)(ox*S);
      float m  = magp[pix];
      int   bb = binp[pix];
      float h[9]; float nr = 0.f;
      #pragma unroll
      for (int o = 0; o < 9; ++o) {
        float hv = spw[o*9 + bb]*m + spb[o];
        h[o] = hv; nr += hv*hv;
      }
      float inv = 1.0f / fmaxf(sqrtf(nr), 1e-12f);
      const float* pp = pred + (((size_t)(n*27 + c*9))*Ho + oy)*Wo + ox;
      #pragma unroll
      for (int o = 0; o < 9; ++o) {
        float d = pp[(size_t)o*Ho*Wo] - h[o]*inv;
        acc += d*d;
      }
    }

    const float* mrow = mask + (size_t)n*PLANE + (size_t)(oy*S)*WW + (size_t)(ox*S);
    float ms = 0.f;
    for (int dy = 0; dy < S; ++dy)
      for (int dx = 0; dx < S; ++dx)
        ms += mrow[(size_t)dy*WW + dx];

    tsum += acc * (1.0f/27.0f) * (ms * (1.0f/(float)(S*S)));
  }

  red[threadIdx.x] = tsum; __syncthreads();
  for (int off = 128; off > 0; off >>= 1) {
    if ((int)threadIdx.x < off) red[threadIdx.x] += red[threadIdx.x + off];
    __syncthreads();
  }
  if (threadIdx.x == 0) partial[blockIdx.x] = red[0];
}

// ---------------- kernel 3b: per-scale fused loss via V_WMMA_F32_16X16X4_F32 ----------------
// D(16x16) = pw_pad(16x12) x HIST(12x16) + pb_pad, chained over three K=4 chunks.
// 16 output pixels per wave per WMMA group; rows M=0..8 of D are h[o] for pixel N.
// Used only for the cheap scales (S=16, 8) where WMMA latency is irrelevant;
// TOT is an exact multiple of 16 for both, so EXEC stays all-ones around WMMA.
#if defined(WMMA_F32_OK)
template<int S>
__global__ void scale_loss_wmma_kernel(const float* __restrict__ magp,
                                       const unsigned char* __restrict__ binp,
                                       const float* __restrict__ mask,
                                       const float* __restrict__ pred,
                                       const float* __restrict__ pw,
                                       const float* __restrict__ pb,
                                       float* __restrict__ partial)
{
  constexpr int Ho = HH / S, Wo = WW / S;
  constexpr int TOT = NIMG * Ho * Wo;
  constexpr int NGROUPS = TOT / 16;            // exact for S=16 (392) and S=8 (1568)

  __shared__ float spw[81];
  __shared__ float spb[9];
  __shared__ float red[256];
  for (int i = threadIdx.x; i < 81; i += blockDim.x) spw[i] = pw[i];
  if (threadIdx.x < 9) spb[threadIdx.x] = pb[threadIdx.x];
  __syncthreads();

  const int lane = threadIdx.x & 31;
  const int wid  = threadIdx.x >> 5;           // wave in block (0..7)
  const int pl   = lane & 15;                  // pixel slot / matrix row
  const int hi   = lane >> 4;                  // half-wave

  // Loop-invariant A operand: pw zero-padded to 16x12.
  // f32 A layout (16x4 per chunk): lanes 0-15 hold K={0,1}, lanes 16-31 K={2,3}.
  v2f a[3];
  #pragma unroll
  for (int kk = 0; kk < 3; ++kk) {
    int k0 = kk*4 + hi*2;
    a[kk].x = (pl < 9 && k0     < 9) ? spw[pl*9 + k0]     : 0.f;
    a[kk].y = (pl < 9 && (k0+1) < 9) ? spw[pl*9 + k0 + 1] : 0.f;
  }
  // Bias baked into C: C[M][N] = pb[M] (M<9) else 0.  C/D layout: vgpr v holds
  // row M=v (lanes 0-15) and M=v+8 (lanes 16-31).
  v8f cinit;
  #pragma unroll
  for (int v = 0; v < 8; ++v)
    cinit[v] = (hi == 0) ? spb[v] : (v == 0 ? spb[8] : 0.f);

  const int nwaves = blockDim.x >> 5;
  float tsum = 0.f;
  for (int g = blockIdx.x*nwaves + wid; g < NGROUPS; g += gridDim.x*nwaves) {
    int q  = g*16 + pl;                        // this lane's pixel (dup across halves)
    int ox = q % Wo; int r = q / Wo;
    int oy = r % Ho; int n = r / Ho;

    float acc = 0.f;
    #pragma unroll
    for (int c = 0; c < 3; ++c) {
      size_t pix = (size_t)(n*3 + c)*PLANE + (size_t)(oy*S)*WW + (size_t)(ox*S);
      float m  = magp[pix];
      int   bb = binp[pix];

      v8f cc = cinit;
      #pragma unroll
      for (int kk = 0; kk < 3; ++kk) {
        // B layout (4x16 per chunk): row K striped across lanes; lanes 0-15 hold
        // K={k0,k0+1} with k0=kk*4, lanes 16-31 hold K={k0+2,k0+3}.
        int k0 = kk*4 + hi*2;
        v2f bv;
        bv.x = (bb == k0    ) ? m : 0.f;       // one-hot column of HIST
        bv.y = (bb == k0 + 1) ? m : 0.f;
        cc = __builtin_amdgcn_wmma_f32_16x16x4_f32(
                 false, a[kk], false, bv, (short)0, cc, false, false);
      }

      // nr = sum over 9 rows; rows 0-7 live in this half, row 8 in the other.
      float nrp = 0.f;
      #pragma unroll
      for (int v = 0; v < 8; ++v) nrp += cc[v]*cc[v];   // padded rows are exactly 0
      float nr  = nrp + __shfl_xor(nrp, 16);
      float inv = 1.0f / fmaxf(sqrtf(nr), 1e-12f);

      const float* pp = pred + (((size_t)(n*27 + c*9 + hi*8))*Ho + oy)*Wo + ox;
      const int rows = hi ? 1 : 8;             // hi=0: o=0..7, hi=1: o=8
      #pragma unroll
      for (int v = 0; v < 8; ++v) {
        if (v < rows) {
          float d = pp[(size_t)v*Ho*Wo] - cc[v]*inv;
          acc += d*d;
        }
      }
    }

    const float* mrow = mask + (size_t)n*PLANE + (size_t)(oy*S)*WW + (size_t)(ox*S);
    float ms = 0.f;
    for (int dy = 0; dy < S; ++dy)
      for (int dx = 0; dx < S; ++dx)
        ms += mrow[(size_t)dy*WW + dx];

    tsum += acc * (1.0f/27.0f) * (ms * (1.0f/(float)(S*S)));
  }

  red[threadIdx.x] = tsum; __syncthreads();
  for (int off = 128; off > 0; off >>= 1) {
    if ((int)threadIdx.x < off) red[threadIdx.x] += red[threadIdx.x + off];
    __syncthreads();
  }
  if (threadIdx.x == 0) partial[blockIdx.x] = red[0];
}
#endif // WMMA_F32_OK

// ---------------- kernel 4: deterministic finalize ----------------
__global__ void finalize_kernel(const float* __restrict__ maskParts,
                                const float* __restrict__ scaleParts,
                                float* __restrict__ out)
{
  __shared__ float red[256];
  __shared__ float msum_sh;
  int t = threadIdx.x;

  float v = 0.f;
  for (int i = t; i < PARTS; i += 256) v += maskParts[i];
  red[t] = v; __syncthreads();
  for (int off = 128; off > 0; off >>= 1) {
    if (t < off) red[t] += red[t + off];
    __syncthreads();
  }
  if (t == 0) msum_sh = red[0];
  __syncthreads();
  float msum = msum_sh;

  const int scales[5] = {16, 8, 4, 2, 1};
  float loss = 0.f;
  for (int k = 0; k < 5; ++k) {
    float s = 0.f;
    for (int i = t; i < PARTS; i += 256) s += scaleParts[k*PARTS + i];
    __syncthreads();
    red[t] = s; __syncthreads();
    for (int off = 128; off > 0; off >>= 1) {
      if (t < off) red[t] += red[t + off];
      __syncthreads();
    }
    if (t == 0) {
      float Msum = msum / (float)(scales[k]*scales[k]);
      loss += red[0] / Msum;
    }
    __syncthreads();
  }
  if (t == 0) out[0] = loss;
}

// ---------------- launch ----------------
extern "C" void kernel_launch(void* const* d_in, const int* in_sizes, int n_in,
                              void* d_out, int out_size, void* d_ws, size_t ws_size,
                              hipStream_t stream)
{
  const float* imgs = (const float*)d_in[0];
  const float* mask = (const float*)d_in[1];
  const float* preds[5]; const float* pws[5]; const float* pbs[5];
  for (int k = 0; k < 5; ++k) {
    preds[k] = (const float*)d_in[2 + k];
    pws[k]   = (const float*)d_in[7 + k];
    pbs[k]   = (const float*)d_in[12 + k];
  }

  char* base = (char*)d_ws;
  float*         magbuf = (float*)base;                       // NPIX floats
  unsigned char* binbuf = (unsigned char*)(base + sizeof(float)*NPIX);
  size_t po = sizeof(float)*NPIX + NPIX;
  po = (po + 255) & ~(size_t)255;
  float* maskParts  = (float*)(base + po);                    // PARTS floats
  float* scaleParts = maskParts + PARTS;                      // 5*PARTS floats

  // 1) per-pixel gradient magnitude + orientation bin (async-LDS tiled Sobel)
  magbin_kernel<<<NPLANES*(HH/TH)*(WW/TW), 256, 0, stream>>>(imgs, magbuf, binbuf);

  // 2) mask total (stage 1)
  mask_reduce_kernel<<<PARTS, 256, 0, stream>>>(mask, maskParts);

  // 3) per-scale fused loss (stage 1)
#if defined(WMMA_F32_OK)
  scale_loss_wmma_kernel<16><<<PARTS, 256, 0, stream>>>(magbuf, binbuf, mask, preds[0], pws[0], pbs[0], scaleParts + 0*PARTS);
  scale_loss_wmma_kernel< 8><<<PARTS, 256, 0, stream>>>(magbuf, binbuf, mask, preds[1], pws[1], pbs[1], scaleParts + 1*PARTS);
#else
  scale_loss_kernel<16><<<PARTS, 256, 0, stream>>>(magbuf, binbuf, mask, preds[0], pws[0], pbs[0], scaleParts + 0*PARTS);
  scale_loss_kernel< 8><<<PARTS, 256, 0, stream>>>(magbuf, binbuf, mask, preds[1], pws[1], pbs[1], scaleParts + 1*PARTS);
#endif
  scale_loss_kernel< 4><<<PARTS, 256, 0, stream>>>(magbuf, binbuf, mask, preds[2], pws[2], pbs[2], scaleParts + 2*PARTS);
  scale_loss_kernel< 2><<<PARTS, 256, 0, stream>>>(magbuf, binbuf, mask, preds[3], pws[3], pbs[3], scaleParts + 3*PARTS);
  scale_loss_kernel< 1><<<PARTS, 256, 0, stream>>>(magbuf, binbuf, mask, preds[4], pws[4], pbs[4], scaleParts + 4*PARTS);

  // 4) deterministic final combine
  finalize_kernel<<<1, 256, 0, stream>>>(maskParts, scaleParts, (float*)d_out);
}